// PrefixSumLinearAttentionModel_33002528703340
// MI455X (gfx1250) — compile-verified
//
#include <hip/hip_runtime.h>
#include <hip/hip_bf16.h>

// Problem constants (from reference)
#define BSZ 32
#define SLEN 2048
#define NVAL 256
#define DD 256
#define VOC 512

typedef __attribute__((ext_vector_type(2))) float v2f;
typedef __attribute__((ext_vector_type(8))) float v8f;

// ---------------------------------------------------------------------------
// Kernel A: M1[key, d] = sum_e embK[key,e] * Wk_w[d, 256+e] + Wk_b[d]
// Dense 512x256x256 f32 GEMM via V_WMMA_F32_16X16X4_F32 (wave32, one wave/tile)
//   A (16x4 f32):  lanes 0-15 hold M=lane, K={k0,k0+1}; lanes 16-31 K={k0+2,k0+3}
//   B (4x16 f32):  VGPR0: rows K=k0(+half), N=lane&15 ; VGPR1: rows K=k0+2(+half)
//   C/D (16x16):   VGPR r -> M = r + 8*half, N = lane&15
// ---------------------------------------------------------------------------
__global__ __launch_bounds__(32) void m1_wmma_kernel(const float* __restrict__ embK,
                                                     const float* __restrict__ Wk_w,
                                                     const float* __restrict__ Wk_b,
                                                     float* __restrict__ M1) {
    int tileM = blockIdx.x >> 4;   // 0..31  (M = 512)
    int tileN = blockIdx.x & 15;   // 0..15  (N = 256)
    int lane  = threadIdx.x;       // 0..31
    int half  = lane >> 4;
    int l16   = lane & 15;
    int mA = tileM * 16 + l16;     // A-matrix row held by this lane
    int nB = tileN * 16 + l16;     // B-matrix column held by this lane

    v8f c = {};
    const float* arow = embK + (size_t)mA * DD;
    const float* bcol = Wk_w + (size_t)nB * VOC + NVAL;  // Wk_w[nB, 256 + e]
    for (int k0 = 0; k0 < DD; k0 += 4) {
        v2f a;
        int ka = k0 + half * 2;
        a.x = arow[ka];
        a.y = arow[ka + 1];
        v2f b;
        b.x = bcol[k0 + half];
        b.y = bcol[k0 + 2 + half];
        c = __builtin_amdgcn_wmma_f32_16x16x4_f32(false, a, false, b,
                                                  (short)0, c, false, false);
    }
    float bias = Wk_b[tileN * 16 + l16];
#pragma unroll
    for (int r = 0; r < 8; ++r) {
        int m = tileM * 16 + half * 8 + r;
        int n = tileN * 16 + l16;
        M1[(size_t)m * DD + n] = c[r] + bias;
    }
}

// ---------------------------------------------------------------------------
// Kernel B: M2[v, d] = Wk_w[d, v]   (transpose of the one-hot half of Wk_w)
// ---------------------------------------------------------------------------
__global__ __launch_bounds__(256) void wk_transpose_kernel(const float* __restrict__ Wk_w,
                                                           float* __restrict__ M2) {
    int v = blockIdx.x;
    int d = threadIdx.x;
    M2[(size_t)v * DD + d] = Wk_w[(size_t)d * VOC + v];
}

// ---------------------------------------------------------------------------
// Kernel C: Q path.  Q[b,d] = embK[q[b]] . Wq_w[d,:] + Wq_b[d]
// Store Qp[b, 0:256] = relu(Q), Qp[b, 256:512] = relu(-Q).
// Also zero-initializes num[] (exactly 32x256 threads in this grid).
// ---------------------------------------------------------------------------
__global__ __launch_bounds__(256) void q_kernel(const int* __restrict__ q,
                                                const float* __restrict__ embK,
                                                const float* __restrict__ Wq_w,
                                                const float* __restrict__ Wq_b,
                                                float* __restrict__ Qp,
                                                float* __restrict__ num) {
    __shared__ float se[DD];
    int b = blockIdx.x;
    int d = threadIdx.x;
    int qi = q[b];
    se[d] = embK[(size_t)qi * DD + d];
    __syncthreads();
    float acc = Wq_b[d];
    const float* wr = Wq_w + (size_t)d * DD;
#pragma unroll 8
    for (int e = 0; e < DD; ++e) acc = fmaf(se[e], wr[e], acc);
    Qp[(size_t)b * 512 + d]       = fmaxf(acc, 0.f);
    Qp[(size_t)b * 512 + 256 + d] = fmaxf(-acc, 0.f);
    num[(size_t)b * NVAL + d] = 0.f;   // zero accumulators for kernel D
}

// ---------------------------------------------------------------------------
// Kernel D: per position (b,l):
//   K_d = M1[key][d] + M2[val][d]
//   w = [relu(K)*Qpos , relu(-K)*Qneg]  (512-dim, circular)
//   s = sum_i w[i]*(w[i-1] + w[i-2])    ( = dpfp(K).dpfp(Q), nu=2 )
//   num[b][val] += s                    (den recovered as sum_v num[b][v])
//
// One wave32 per position. Lane L owns contiguous chunks wpos[8L..8L+7] and
// wneg[8L..8L+7] in REGISTERS; the circular shift only needs elements 6,7 of
// the previous lane's chunks -> 4 cross-lane shuffles, zero LDS traffic.
// Lane 0 handles the circular wrap by swapping the shuffled pos/neg pairs.
// ---------------------------------------------------------------------------
__global__ __launch_bounds__(256) void attn_kernel(const int* __restrict__ x,
                                                   const float* __restrict__ M1,
                                                   const float* __restrict__ M2,
                                                   const float* __restrict__ Qp,
                                                   float* __restrict__ num) {
    int b     = blockIdx.x >> 5;   // 32 blocks per batch row
    int chunk = blockIdx.x & 31;   // 64 positions per block
    int tid   = threadIdx.x;
    int wave  = tid >> 5;
    int lane  = tid & 31;
    int prevl = (lane + 31) & 31;  // lane-1 with wrap

    // Per-lane Q' registers: Qpos[8L..8L+7], Qneg[8L..8L+7]
    const float4* qpv = (const float4*)(Qp + (size_t)b * 512 + lane * 8);
    const float4* qnv = (const float4*)(Qp + (size_t)b * 512 + 256 + lane * 8);
    float4 qp0 = qpv[0], qp1 = qpv[1];
    float4 qn0 = qnv[0], qn1 = qnv[1];
    float qp[8] = {qp0.x, qp0.y, qp0.z, qp0.w, qp1.x, qp1.y, qp1.z, qp1.w};
    float qn[8] = {qn0.x, qn0.y, qn0.z, qn0.w, qn1.x, qn1.y, qn1.z, qn1.w};

    const int* keyp = x + (size_t)b * 2 * SLEN;
    const int* valp = keyp + SLEN;
    int lbase = chunk * 64 + wave * 8;

    for (int t = 0; t < 8; ++t) {
        int l   = lbase + t;
        int key = keyp[l];
        int val = valp[l] - NVAL;
        if (t < 7) {  // prefetch next position's table rows (global_prefetch_b8)
            int kn = keyp[l + 1], vn = valp[l + 1] - NVAL;
            __builtin_prefetch(M1 + (size_t)kn * DD + lane * 8, 0, 1);
            __builtin_prefetch(M2 + (size_t)vn * DD + lane * 8, 0, 1);
        }
        // coalesced b128 loads: lane L reads d = 8L..8L+7 of each table row
        const float4* m1v = (const float4*)(M1 + (size_t)key * DD + lane * 8);
        const float4* m2v = (const float4*)(M2 + (size_t)val * DD + lane * 8);
        float4 a0 = m1v[0], a1 = m1v[1];
        float4 b0 = m2v[0], b1 = m2v[1];
        float kd[8] = {a0.x + b0.x, a0.y + b0.y, a0.z + b0.z, a0.w + b0.w,
                       a1.x + b1.x, a1.y + b1.y, a1.z + b1.z, a1.w + b1.w};
        float wp[8], wn[8];
#pragma unroll
        for (int j = 0; j < 8; ++j) {
            wp[j] = fmaxf(kd[j], 0.f)  * qp[j];
            wn[j] = fmaxf(-kd[j], 0.f) * qn[j];
        }
        // boundary elements from previous lane (wrap lane0 <- lane31)
        float pp6 = __shfl(wp[6], prevl, 32);
        float pp7 = __shfl(wp[7], prevl, 32);
        float pn6 = __shfl(wn[6], prevl, 32);
        float pn7 = __shfl(wn[7], prevl, 32);
        bool first = (lane == 0);
        // prev elems for pos-chunk: normally prev lane's wpos; lane0 wraps to wneg[254,255]
        float pa1 = first ? pn7 : pp7;   // w[i-1] for j=0
        float pa2 = first ? pn6 : pp6;   // w[i-2] for j=0
        // prev elems for neg-chunk: normally prev lane's wneg; lane0 wraps to wpos[254,255]
        float na1 = first ? pp7 : pn7;
        float na2 = first ? pp6 : pn6;

        float s;
        s = wp[0] * (pa1 + pa2);
        s = fmaf(wp[1], wp[0] + pa1, s);
#pragma unroll
        for (int j = 2; j < 8; ++j) s = fmaf(wp[j], wp[j - 1] + wp[j - 2], s);
        s = fmaf(wn[0], na1 + na2, s);
        s = fmaf(wn[1], wn[0] + na1, s);
#pragma unroll
        for (int j = 2; j < 8; ++j) s = fmaf(wn[j], wn[j - 1] + wn[j - 2], s);

        // wave32 butterfly reduction
#pragma unroll
        for (int off = 16; off > 0; off >>= 1) s += __shfl_xor(s, off, 32);
        if (lane == 0) atomicAdd(&num[(size_t)b * NVAL + val], s);
    }
}

// ---------------------------------------------------------------------------
// Kernel E: y[b,v] = num[b,v] / (sum_v num[b,v] + eps)
// ---------------------------------------------------------------------------
__global__ __launch_bounds__(256) void finalize_kernel(const float* __restrict__ num,
                                                       float* __restrict__ y) {
    __shared__ float red[256];
    int b = blockIdx.x;
    int v = threadIdx.x;
    float s = num[(size_t)b * NVAL + v];
    red[v] = s;
    __syncthreads();
#pragma unroll
    for (int off = 128; off > 0; off >>= 1) {
        if (v < off) red[v] += red[v + off];
        __syncthreads();
    }
    float den = red[0] + 1e-6f;
    y[(size_t)b * NVAL + v] = s / den;
}

// ---------------------------------------------------------------------------
extern "C" void kernel_launch(void* const* d_in, const int* in_sizes, int n_in,
                              void* d_out, int out_size, void* d_ws, size_t ws_size,
                              hipStream_t stream) {
    const int*   x    = (const int*)d_in[0];    // (32, 2, 2048) int32
    const int*   q    = (const int*)d_in[1];    // (32, 1) int32
    const float* embK = (const float*)d_in[2];  // (512, 256)
    const float* Wk_w = (const float*)d_in[3];  // (256, 512)
    const float* Wk_b = (const float*)d_in[4];  // (256,)
    const float* Wq_w = (const float*)d_in[5];  // (256, 256)
    const float* Wq_b = (const float*)d_in[6];  // (256,)
    float* y = (float*)d_out;                   // (32, 256)

    float* M1  = (float*)d_ws;          // 512*256 f32  (embK @ Wk2^T + bias)
    float* M2  = M1 + VOC * DD;         // 256*256 f32  (Wk one-hot half, transposed)
    float* Qp  = M2 + NVAL * DD;        // 32*512  f32  ([relu(Q), relu(-Q)])
    float* num = Qp + BSZ * 512;        // 32*256  f32  (per-value accumulators)

    m1_wmma_kernel<<<512, 32, 0, stream>>>(embK, Wk_w, Wk_b, M1);
    wk_transpose_kernel<<<NVAL, 256, 0, stream>>>(Wk_w, M2);
    q_kernel<<<BSZ, 256, 0, stream>>>(q, embK, Wq_w, Wq_b, Qp, num);
    attn_kernel<<<BSZ * 32, 256, 0, stream>>>(x, M1, M2, Qp, num);
    finalize_kernel<<<BSZ, 256, 0, stream>>>(num, y);
}